// GAT_38551626449703
// MI455X (gfx1250) — compile-verified
//
#include <hip/hip_runtime.h>
#include <cstdint>
#include <cstddef>

typedef float v2f __attribute__((ext_vector_type(2)));
typedef float v8f __attribute__((ext_vector_type(8)));

#define GAT_ALPHA 0.2f

__device__ __forceinline__ void atomicMaxF(float* addr, float v) {
    // Monotonic int mapping trick: works for mixed-sign floats (init to -inf).
    if (v >= 0.0f) atomicMax((int*)addr, __float_as_int(v));
    else           atomicMin((unsigned int*)addr, __float_as_uint(v));
}

__device__ __forceinline__ void atomAddF(float* p, float v) {
#if defined(__HIP_PLATFORM_AMD__)
    unsafeAtomicAdd(p, v);   // native global_atomic_add_f32, no CAS loop
#else
    atomicAdd(p, v);
#endif
}

__global__ void k_fill(float* __restrict__ p, float v, int n) {
    int i = blockIdx.x * blockDim.x + threadIdx.x;
    if (i < n) p[i] = v;
}

// C[b][M x 64] = A[b][M x K] @ B[b][K x 64] using V_WMMA_F32_16X16X4_F32.
// One wave per 16x64 output tile; grid = (M/16, batch). M % 16 == 0, K % 4 == 0.
__global__ __launch_bounds__(32) void k_gemm_wmma(
    const float* __restrict__ A, const float* __restrict__ B, float* __restrict__ C,
    int K, long aStride, long bStride, long cStride)
{
    const int lane = threadIdx.x;          // 0..31, EXEC all ones (required by WMMA)
    const int m0   = blockIdx.x * 16;
    const int b    = blockIdx.y;
    const float* __restrict__ Ab = A + (long)b * aStride;
    const float* __restrict__ Bb = B + (long)b * bStride;
    float* __restrict__ Cb = C + (long)b * cStride;

    const int row = lane & 15;             // A: M = lane&15 (both halves)
    const int kh  = (lane >> 4) << 1;      // A/B: K sub-offset 0 (lanes 0-15) or 2 (lanes 16-31)
    const int col = lane & 15;             // B/C/D: N = lane&15

    v8f c0 = {}, c1 = {}, c2 = {}, c3 = {};

    for (int k0 = 0; k0 < K; k0 += 4) {
        // A 16x4 tile: VGPR0 = K=kh, VGPR1 = K=kh+1 (contiguous pair -> b64 load)
        const float2 af = *reinterpret_cast<const float2*>(
            Ab + (size_t)(m0 + row) * K + k0 + kh);
        v2f a; a.x = af.x; a.y = af.y;

        // B 4x16 tiles for n0 = 0,16,32,48: VGPR0 = row K=k0+kh, VGPR1 = row K=k0+kh+1
        const float* br0 = Bb + (size_t)(k0 + kh) * 64 + col;
        const float* br1 = br0 + 64;
        v2f b0, b1, b2, b3;
        b0.x = br0[0];  b0.y = br1[0];
        b1.x = br0[16]; b1.y = br1[16];
        b2.x = br0[32]; b2.y = br1[32];
        b3.x = br0[48]; b3.y = br1[48];

        c0 = __builtin_amdgcn_wmma_f32_16x16x4_f32(false, a, false, b0, (short)0, c0, false, false);
        c1 = __builtin_amdgcn_wmma_f32_16x16x4_f32(false, a, false, b1, (short)0, c1, false, false);
        c2 = __builtin_amdgcn_wmma_f32_16x16x4_f32(false, a, false, b2, (short)0, c2, false, false);
        c3 = __builtin_amdgcn_wmma_f32_16x16x4_f32(false, a, false, b3, (short)0, c3, false, false);
    }

    // D layout: VGPR r -> (M = r, N = lane) for lanes 0-15, (M = 8+r, N = lane-16) for 16-31
    const int rbase = m0 + ((lane >> 4) << 3);
    #pragma unroll
    for (int r = 0; r < 8; ++r) {
        float* crow = Cb + (size_t)(rbase + r) * 64 + col;
        crow[0]  = c0[r];
        crow[16] = c1[r];
        crow[32] = c2[r];
        crow[48] = c3[r];
    }
}

// Per (node,head) wave: s_src = Wh[n] . a[h][0:64], s_dst = Wh[n] . a[h][64:128]
__global__ void k_node_scores(const float* __restrict__ Wh, const float* __restrict__ a,
                              float* __restrict__ ssrc, float* __restrict__ sdst,
                              int NH, int N)
{
    int wid  = (blockIdx.x * blockDim.x + threadIdx.x) >> 5;
    int lane = threadIdx.x & 31;
    if (wid >= NH) return;
    int h = wid / N;
    const float* __restrict__ row = Wh + (size_t)wid * 64;
    const float* __restrict__ as  = a + h * 128;
    const float* __restrict__ ad  = as + 64;
    float w0 = row[lane], w1 = row[lane + 32];
    float s0 = w0 * as[lane] + w1 * as[lane + 32];
    float s1 = w0 * ad[lane] + w1 * ad[lane + 32];
    #pragma unroll
    for (int off = 16; off >= 1; off >>= 1) {
        s0 += __shfl_xor(s0, off, 32);
        s1 += __shfl_xor(s1, off, 32);
    }
    if (lane == 0) { ssrc[wid] = s0; sdst[wid] = s1; }
}

// e = leakyrelu(ssrc[src] + sdst[dst]); segment max over src via float atomic max
__global__ void k_edge_max(const int* __restrict__ src, const int* __restrict__ dst,
                           const float* __restrict__ ssrc, const float* __restrict__ sdst,
                           float* __restrict__ ebuf, float* __restrict__ m,
                           int E, int H, int N)
{
    int tid = blockIdx.x * blockDim.x + threadIdx.x;
    if (tid >= E * H) return;
    int e = tid % E, h = tid / E;
    int s = src[e], d = dst[e];
    float v = ssrc[h * N + s] + sdst[h * N + d];
    v = v > 0.0f ? v : GAT_ALPHA * v;
    ebuf[tid] = v;
    atomicMaxF(&m[h * N + s], v);
}

// ex = exp(e - m[src]); segment sum of ex
__global__ void k_edge_exp(const int* __restrict__ src, const float* __restrict__ m,
                           float* __restrict__ ebuf, float* __restrict__ denom,
                           int E, int H, int N)
{
    int tid = blockIdx.x * blockDim.x + threadIdx.x;
    if (tid >= E * H) return;
    int e = tid % E, h = tid / E;
    int s = src[e];
    float ex = __expf(ebuf[tid] - m[h * N + s]);
    ebuf[tid] = ex;
    atomAddF(&denom[h * N + s], ex);
}

// h'[src] += att * Wh[dst]; wave per (edge,head), 2 dims per lane
__global__ void k_edge_scatter(const int* __restrict__ src, const int* __restrict__ dst,
                               const float* __restrict__ ebuf, const float* __restrict__ denom,
                               const float* __restrict__ Wh, float* __restrict__ hp,
                               int E, int H, int N)
{
    int wid  = (blockIdx.x * blockDim.x + threadIdx.x) >> 5;
    int lane = threadIdx.x & 31;
    if (wid >= E * H) return;
    int e = wid % E, h = wid / E;
    int s = src[e], d = dst[e];
    float att = ebuf[wid] / denom[h * N + s];
    const float* __restrict__ wrow = Wh + (size_t)(h * N + d) * 64;
    float* __restrict__ orow = hp + (size_t)(h * N + s) * 64;
    atomAddF(&orow[lane],      att * wrow[lane]);
    atomAddF(&orow[lane + 32], att * wrow[lane + 32]);
}

// hcat[n][h*64+d] = elu(hp[h][n][d])
__global__ void k_elu_concat(const float* __restrict__ hp, float* __restrict__ hcat, int N) {
    int tid = blockIdx.x * blockDim.x + threadIdx.x;
    if (tid >= N * 256) return;
    int n = tid >> 8, c = tid & 255, h = c >> 6, d = c & 63;
    float v = hp[((size_t)h * N + n) * 64 + d];
    hcat[tid] = v > 0.0f ? v : __expf(v) - 1.0f;
}

__global__ void k_elu(const float* __restrict__ in, float* __restrict__ out, int n) {
    int i = blockIdx.x * blockDim.x + threadIdx.x;
    if (i < n) {
        float v = in[i];
        out[i] = v > 0.0f ? v : __expf(v) - 1.0f;
    }
}

static inline int cdiv(long a, int b) { return (int)((a + b - 1) / b); }

extern "C" void kernel_launch(void* const* d_in, const int* in_sizes, int n_in,
                              void* d_out, int out_size, void* d_ws, size_t ws_size,
                              hipStream_t stream)
{
    const float* x       = (const float*)d_in[0];   // [N,128]
    const int*   edges   = (const int*)d_in[1];     // [2,E]
    const float* W_heads = (const float*)d_in[2];   // [4,128,64]
    const float* a_heads = (const float*)d_in[3];   // [4,128]
    const float* W_out   = (const float*)d_in[4];   // [256,64]
    const float* a_out   = (const float*)d_in[5];   // [128]
    float* out = (float*)d_out;                     // [N,64]

    const int N = in_sizes[0] / 128;
    const int E = in_sizes[1] / 2;
    const int H = 4, HID = 64;

    const int* src = edges;
    const int* dst = edges + E;

    // ---- workspace layout (floats) ----
    float* ws = (float*)d_ws;
    size_t off = 0;
    float* Wh1   = ws + off; off += (size_t)H * N * HID;   // 12.8M
    float* hcat  = ws + off; off += (size_t)N * 256;       // 12.8M
    float* Wh2   = ws + off; off += (size_t)N * HID;       // 3.2M
    float* ssrc1 = ws + off; off += (size_t)H * N;
    float* sdst1 = ws + off; off += (size_t)H * N;
    float* ssrc2 = ws + off; off += (size_t)N;
    float* sdst2 = ws + off; off += (size_t)N;
    float* ebuf1 = ws + off; off += (size_t)H * E;
    float* ebuf2 = ws + off; off += (size_t)E;
    // contiguous zero-init region
    float* zbase = ws + off;
    float* denom1 = ws + off; off += (size_t)H * N;
    float* hp1    = ws + off; off += (size_t)H * N * HID;
    float* denom2 = ws + off; off += (size_t)N;
    float* acc2   = ws + off; off += (size_t)N * HID;
    size_t zcount = (size_t)(ws + off - zbase);
    // contiguous -inf region
    float* nbase = ws + off;
    float* m1 = ws + off; off += (size_t)H * N;
    float* m2 = ws + off; off += (size_t)N;
    size_t ncount = (size_t)(ws + off - nbase);

    const int BS = 256;

    // init
    k_fill<<<cdiv((long)zcount, BS), BS, 0, stream>>>(zbase, 0.0f, (int)zcount);
    k_fill<<<cdiv((long)ncount, BS), BS, 0, stream>>>(nbase, -__builtin_huge_valf(), (int)ncount);

    // ---- layer 1 ----
    // Wh1[h] = x @ W_heads[h]   (A stride 0: same x for all heads)
    k_gemm_wmma<<<dim3(N / 16, H), 32, 0, stream>>>(x, W_heads, Wh1,
        128, 0L, (long)128 * 64, (long)N * 64);
    // per-node attention scores
    k_node_scores<<<cdiv((long)H * N * 32, BS), BS, 0, stream>>>(Wh1, a_heads, ssrc1, sdst1, H * N, N);
    // edge softmax (max -> exp/sum -> normalize+scatter)
    k_edge_max<<<cdiv((long)E * H, BS), BS, 0, stream>>>(src, dst, ssrc1, sdst1, ebuf1, m1, E, H, N);
    k_edge_exp<<<cdiv((long)E * H, BS), BS, 0, stream>>>(src, m1, ebuf1, denom1, E, H, N);
    k_edge_scatter<<<cdiv((long)E * H * 32, BS), BS, 0, stream>>>(src, dst, ebuf1, denom1, Wh1, hp1, E, H, N);
    // elu + head concat
    k_elu_concat<<<cdiv((long)N * 256, BS), BS, 0, stream>>>(hp1, hcat, N);

    // ---- layer 2 ----
    k_gemm_wmma<<<dim3(N / 16, 1), 32, 0, stream>>>(hcat, W_out, Wh2,
        256, 0L, 0L, 0L);
    k_node_scores<<<cdiv((long)N * 32, BS), BS, 0, stream>>>(Wh2, a_out, ssrc2, sdst2, N, N);
    k_edge_max<<<cdiv((long)E, BS), BS, 0, stream>>>(src, dst, ssrc2, sdst2, ebuf2, m2, E, 1, N);
    k_edge_exp<<<cdiv((long)E, BS), BS, 0, stream>>>(src, m2, ebuf2, denom2, E, 1, N);
    k_edge_scatter<<<cdiv((long)E * 32, BS), BS, 0, stream>>>(src, dst, ebuf2, denom2, Wh2, acc2, E, 1, N);

    // final elu -> d_out
    k_elu<<<cdiv((long)N * HID, BS), BS, 0, stream>>>(acc2, out, N * HID);
}